// CustomReshapeLayer_26792005992501
// MI455X (gfx1250) — compile-verified
//
#include <hip/hip_runtime.h>
#include <cstdint>
#include <cstddef>

// Staircase gather: out[b, i, j] = in[b, i*M - i*(i-1)/2 + j]
// M = 2048, B = 32.  Pure data movement, HBM-store-bound (~537 MB out,
// reads ~L2-resident).  One 256-thread block per output row; stage the row
// through LDS using the CDNA5 async data-mover (ASYNCcnt): B32 async loads
// (source rows only 4B-aligned) and non-temporal B128 async stores (dest
// rows 8KB-aligned; output is streaming, keep it out of L2).

#define MATSIZE 2048u
#define BATCH   32u
#define VEC_LEN (MATSIZE * (MATSIZE + 1u) / 2u)   // 2,098,176
#define ROWS    (BATCH * MATSIZE)                  // 65,536 blocks
#define TPB     256u

// cpol temporal-hint bits (gfx1250): TH in [2:0]; store TH=1 -> NT
#define CPOL_RT 0
#define CPOL_NT 1

// ---- CDNA5 async global<->LDS path, guarded so the file always compiles ----
#if defined(__HIP_DEVICE_COMPILE__) && defined(__gfx1250__) &&                  \
    __has_builtin(__builtin_amdgcn_global_load_async_to_lds_b32) &&             \
    __has_builtin(__builtin_amdgcn_global_store_async_from_lds_b128)
#define USE_CDNA5_ASYNC 1
#else
#define USE_CDNA5_ASYNC 0
#endif

__device__ __forceinline__ void wait_asynccnt0() {
#if defined(__HIP_DEVICE_COMPILE__) && __has_builtin(__builtin_amdgcn_s_wait_asynccnt)
  __builtin_amdgcn_s_wait_asynccnt(0);
#elif defined(__HIP_DEVICE_COMPILE__)
  asm volatile("s_wait_asynccnt 0" ::: "memory");
#endif
}

__global__ __launch_bounds__(TPB) void staircase_gather_kernel(
    const float* __restrict__ in, float* __restrict__ out) {
  const uint32_t r = blockIdx.x;              // 0 .. ROWS-1
  const uint32_t b = r >> 11;                 // r / 2048
  const uint32_t i = r & (MATSIZE - 1u);      // r % 2048
  const uint32_t start = i * MATSIZE - (i * (i - 1u)) / 2u;  // max ~2.1M
  const float* __restrict__ src = in + (size_t)b * VEC_LEN + start;
  float* __restrict__ dst       = out + (size_t)r * MATSIZE;
  const uint32_t t = threadIdx.x;

#if USE_CDNA5_ASYNC
  __shared__ float lbuf[MATSIZE];             // 8 KB row staging buffer

  // Typed, addrspace-qualified pointers per the builtin signatures:
  //   b32 : (as1 int*,  as3 int*,  imm offset, imm cpol)
  //   b128: (as1 v4i*,  as3 v4i*,  imm offset, imm cpol)
  typedef int v4i __attribute__((vector_size(16)));
  typedef __attribute__((address_space(1))) int  gint_t;
  typedef __attribute__((address_space(3))) int  lint_t;
  typedef __attribute__((address_space(1))) v4i  gv4i_t;
  typedef __attribute__((address_space(3))) v4i  lv4i_t;

  // Phase 1: global -> LDS, 4B granularity (source only 4B-aligned).
  // 2048 elems / 256 threads = 8 async B32 loads per thread.  RT hint:
  // input has ~2x reuse and a tiny (8.4 MB/batch) footprint -> L2-resident.
#pragma unroll
  for (uint32_t k = 0; k < MATSIZE / TPB; ++k) {
    const uint32_t e = t + TPB * k;
    __builtin_amdgcn_global_load_async_to_lds_b32(
        (gint_t*)(src + e), (lint_t*)(lbuf + e), /*offset=*/0, CPOL_RT);
  }
  wait_asynccnt0();      // our wave's async loads landed in LDS
  __syncthreads();       // all waves' loads landed

  // Phase 2: LDS -> global, 16B granularity (dest row is 8KB-aligned).
  // 8192 bytes / (256 threads * 16B) = 2 async B128 stores per thread.
  // NT hint: 537 MB streaming output, never re-read -> don't thrash L2.
#pragma unroll
  for (uint32_t k = 0; k < (MATSIZE * 4u) / (TPB * 16u); ++k) {
    const uint32_t byteoff = (t + TPB * k) * 16u;
    __builtin_amdgcn_global_store_async_from_lds_b128(
        (gv4i_t*)((char*)dst + byteoff),
        (lv4i_t*)((char*)lbuf + byteoff), /*offset=*/0, CPOL_NT);
  }
  wait_asynccnt0();      // (S_ENDPGM also implies wait-idle; explicit is safer)
#else
  // Fallback: coalesced copy, scalar loads (4B-aligned source), float4 stores.
#pragma unroll
  for (uint32_t k = 0; k < (MATSIZE / 4u) / TPB; ++k) {
    const uint32_t e = 4u * (t + TPB * k);
    float4 v;
    v.x = src[e + 0]; v.y = src[e + 1]; v.z = src[e + 2]; v.w = src[e + 3];
    *(float4*)(dst + e) = v;
  }
#endif
}

extern "C" void kernel_launch(void* const* d_in, const int* in_sizes, int n_in,
                              void* d_out, int out_size, void* d_ws, size_t ws_size,
                              hipStream_t stream) {
  (void)in_sizes; (void)n_in; (void)out_size; (void)d_ws; (void)ws_size;
  const float* in = (const float*)d_in[0];
  float* out      = (float*)d_out;
  staircase_gather_kernel<<<dim3(ROWS), dim3(TPB), 0, stream>>>(in, out);
}